// FraunhoferPILoss_69982197121524
// MI455X (gfx1250) — compile-verified
//
#include <hip/hip_runtime.h>
#include <math.h>

// ---------------- problem constants ----------------
#define MASKP   512
#define NS      2048           // N_SAMPLES
#define NPTS    4
#define NBATCH  2
#define TPB     256
#define NBLOCKS 4096
#define TOTAL_T (NBLOCKS * TPB)          // 1,048,576 threads
#define CELLS_B (NS * NS)                // 4,194,304 cells per batch
#define ITERS   (CELLS_B / TOTAL_T)      // 4

#define DXF   6.0e-6f
#define EPSF  2.2204e-16f
#define STEPF (2048.0f / 2047.0f)
// K = 2*pi/3.55e-7, computed in double then narrowed, matching numpy
#define KF    ((float)(6.283185307179586 / 3.55e-7))

typedef __attribute__((ext_vector_type(2))) float v2f;
typedef __attribute__((ext_vector_type(8))) float v8f;

// Numerical-Recipes J1, branchless clamped branches (mirrors reference)
__device__ __forceinline__ float j1_nr(float x) {
    float ax = fabsf(x);
    // small branch (|x|<8), clamped input
    float xc = fminf(fmaxf(x, -8.0f), 8.0f);
    float y  = xc * xc;
    float num = xc * (72362614232.0f + y * (-7895059235.0f + y * (242396853.1f +
                y * (-2972611.439f + y * (15704.48260f + y * (-30.16036606f))))));
    float den = 144725228442.0f + y * (2300535178.0f + y * (18583304.74f +
                y * (99447.43394f + y * (376.9991397f + y))));
    float small_v = num / den;
    // large branch (|x|>=8), clamped input
    float axl = fmaxf(ax, 8.0f);
    float z   = 8.0f / axl;
    float y2  = z * z;
    float xx  = axl - 2.356194491f;
    float p1  = 1.0f + y2 * (0.183105e-2f + y2 * (-0.3516396496e-4f +
                y2 * (0.2457520174e-5f + y2 * (-0.240337019e-6f))));
    float p2  = 0.04687499995f + y2 * (-0.2002690873e-3f + y2 * (0.8449199096e-5f +
                y2 * (-0.88228987e-6f + y2 * 0.105787412e-6f)));
    float s, c;
    sincosf(xx, &s, &c);
    float sgn = (x > 0.0f) ? 1.0f : ((x < 0.0f) ? -1.0f : 0.0f);
    float large_v = sqrtf(0.636619772f / axl) * (c * p1 - z * s * p2) * sgn;
    return (ax < 8.0f) ? small_v : large_v;
}

// ---- kernel 1: gather per-point parameters (16 combos = hyp*8 + b*4 + p) ----
// prm[c*5 + {0..4}] = xm, ym, rs_/2, K*rs_/zs_, K/(2*zs_)
__global__ void fh_setup(const float* __restrict__ y_hat, const float* __restrict__ y,
                         const int* __restrict__ xs, const int* __restrict__ ys,
                         float* __restrict__ prm) {
    int c = threadIdx.x;
    if (c >= 16) return;
    int hyp = c >> 3;
    int b   = (c >> 2) & 1;
    int p   = c & 3;
    int xi = xs[b * NPTS + p];
    int yi = ys[b * NPTS + p];
    const int plane = MASKP * MASKP;
    float rs, zs;
    if (hyp == 0) {
        rs = 50.0f  * y_hat[(b * 3 + 2) * plane + yi * MASKP + xi];
        zs = 200.0f * y_hat[(b * 3 + 1) * plane + yi * MASKP + xi];
    } else {
        rs = y[(b * 5 + 2) * plane + yi * MASKP + xi];
        zs = y[(b * 5 + 1) * plane + yi * MASKP + xi];
    }
    float rs_ = rs * 1.0e-6f;   // * DR
    float zs_ = zs * 1.0e-3f;   // * DZ
    prm[c * 5 + 0] = (float)(xi * 4 - 1024) * DXF;   // xm
    prm[c * 5 + 1] = (float)(yi * 4 - 1024) * DXF;   // ym
    prm[c * 5 + 2] = 0.5f * rs_;                     // rs_/2
    prm[c * 5 + 3] = (KF * rs_) / zs_;               // J1 arg multiplier
    prm[c * 5 + 4] = KF / (2.0f * zs_);              // phase multiplier
}

// ---- kernel 2: hologram synthesis + squared-diff partial reduction ----
__global__ void __launch_bounds__(TPB)
fh_main(const float* __restrict__ prm, float* __restrict__ partial) {
    const int g = blockIdx.x * TPB + threadIdx.x;
    float acc = 0.0f;
    for (int b = 0; b < NBATCH; ++b) {
        float xm[8], ym[8], hr[8], a1[8], cp[8];
#pragma unroll
        for (int q = 0; q < 8; ++q) {
            int hyp = q >> 2, p = q & 3;
            int c = hyp * 8 + b * 4 + p;
            xm[q] = prm[c * 5 + 0];
            ym[q] = prm[c * 5 + 1];
            hr[q] = prm[c * 5 + 2];
            a1[q] = prm[c * 5 + 3];
            cp[q] = prm[c * 5 + 4];
        }
        for (int it = 0; it < ITERS; ++it) {
            int idx = g + it * TOTAL_T;
            int i = idx >> 11;       // row (y axis)
            int j = idx & (NS - 1);  // col (x axis)
            float ny = fmaf((float)i, STEPF, -1024.0f) * DXF;
            float nx = fmaf((float)j, STEPF, -1024.0f) * DXF;
            float vals[2];
#pragma unroll
            for (int hyp = 0; hyp < 2; ++hyp) {
                float sr = 0.0f, si = 0.0f;
#pragma unroll
                for (int p = 0; p < 4; ++p) {
                    int q = hyp * 4 + p;
                    float dx = (nx - xm[q]) + EPSF;
                    float dy = (ny - ym[q]) + EPSF;
                    float dx2 = dx * dx;
                    float dy2 = dy * dy;
                    float r2  = dx2 + dy2;
                    float rho = sqrtf(r2);
                    float j1v = j1_nr(a1[q] * rho);
                    float amp = (hr[q] / rho) * j1v;
                    float ph  = cp[q] * (rho * rho);
                    float s, c;
                    sincosf(ph, &s, &c);
                    sr = fmaf(amp, s, sr);
                    si = fmaf(amp, c, si);
                }
                float re = sr - 1.0f;
                float im = -si;
                vals[hyp] = re * re + im * im;
            }
            float d = vals[0] - vals[1];
            acc = fmaf(d, d, acc);
        }
    }
    // deterministic block tree reduction
    __shared__ float red[TPB];
    red[threadIdx.x] = acc;
    __syncthreads();
    for (int s = TPB / 2; s > 0; s >>= 1) {
        if (threadIdx.x < s) red[threadIdx.x] += red[threadIdx.x + s];
        __syncthreads();
    }
    if (threadIdx.x == 0) partial[blockIdx.x] = red[0];
}

// ---- kernel 3: final reduction via V_WMMA_F32_16X16X4_F32 (B = ones trick) ----
// A (16x4 f32) occupies exactly 2 VGPRs x 32 lanes bijectively, so 4 chained
// WMMAs consume all 256 pre-summed values; B=ones => C columns replicate row
// sums, so sum(all C slots) == 16 * total (exact /16 rescale afterwards).
__global__ void fh_final(const float* __restrict__ partial, float* __restrict__ out) {
    __shared__ float sm[TPB];
    int t = threadIdx.x;
    float s = 0.0f;
    for (int k = t; k < NBLOCKS; k += TPB) s += partial[k];  // fixed order: deterministic
    sm[t] = s;
    __syncthreads();
    if (t < 32) {
        v8f cacc = {};
        v2f bones;
        bones.x = 1.0f;
        bones.y = 1.0f;
#pragma unroll
        for (int chunk = 0; chunk < 4; ++chunk) {
            v2f a;
            a.x = sm[chunk * 64 + t];
            a.y = sm[chunk * 64 + 32 + t];
            cacc = __builtin_amdgcn_wmma_f32_16x16x4_f32(
                /*neg_a=*/false, a, /*neg_b=*/false, bones,
                /*c_mod=*/(short)0, cacc, /*reuse_a=*/false, /*reuse_b=*/false);
        }
        float lc = cacc[0] + cacc[1] + cacc[2] + cacc[3] +
                   cacc[4] + cacc[5] + cacc[6] + cacc[7];
        // butterfly over 32 lanes -> every lane holds 16 * total
        for (int off = 16; off > 0; off >>= 1)
            lc += __shfl_xor(lc, off, 32);
        if (t == 0) {
            // total = lc/16 ; result = mean * N^2/384^2 = total / (B*384^2)
            out[0] = lc * (1.0f / 16.0f) * (1.0f / 294912.0f);
        }
    }
}

extern "C" void kernel_launch(void* const* d_in, const int* in_sizes, int n_in,
                              void* d_out, int out_size, void* d_ws, size_t ws_size,
                              hipStream_t stream) {
    const float* y_hat = (const float*)d_in[0];  // (2,3,512,512)
    const float* y     = (const float*)d_in[1];  // (2,5,512,512)
    const int*   xs    = (const int*)d_in[2];    // (2,4)
    const int*   ys    = (const int*)d_in[3];    // (2,4)
    float* out = (float*)d_out;

    float* prm     = (float*)d_ws;               // 80 floats
    float* partial = (float*)d_ws + 512;         // 4096 floats (2KB offset keeps alignment)

    fh_setup<<<1, 32, 0, stream>>>(y_hat, y, xs, ys, prm);
    fh_main<<<NBLOCKS, TPB, 0, stream>>>(prm, partial);
    fh_final<<<1, TPB, 0, stream>>>(partial, out);
}